// RBERT_73126113182287
// MI455X (gfx1250) — compile-verified
//
#include <hip/hip_runtime.h>

#define B_   128
#define S_   2048
#define H_   768
#define L_   29
#define K3H  (3 * H_)              // 2304
#define CHUNK_ROWS 64
#define NCHUNK (S_ / CHUNK_ROWS)   // 32

typedef __attribute__((ext_vector_type(2))) float v2f;
typedef __attribute__((ext_vector_type(4))) float v4f;
typedef __attribute__((ext_vector_type(8))) float v8f;

// ---------------------------------------------------------------- zero ws ---
__global__ __launch_bounds__(256) void rbert_zero(float* __restrict__ p, int n) {
    int i = blockIdx.x * 256 + threadIdx.x;
    if (i < n) p[i] = 0.0f;
}

// ------------------------------------------------- span partial sums (BW) ---
// grid: (NCHUNK, B*2), block: 192 threads, thread t owns channels 4t..4t+3.
// Streams up to 64 rows x 3KB per block; 4-row unroll for load-level
// parallelism; non-temporal b128 loads keep the 805MB 'hidden' stream out
// of the 192MB L2 so the small hot data (accum/x/W) stays resident.
__global__ __launch_bounds__(192) void rbert_span_partial(
    const float* __restrict__ hidden,
    const int*   __restrict__ e1,
    const int*   __restrict__ e2,
    float*       __restrict__ accum)   // [B*2, H]
{
    const int chunk = blockIdx.x;
    const int be    = blockIdx.y;       // b*2 + ent
    const int b     = be >> 1;
    const int ent   = be & 1;
    const int* pos  = ent ? e2 : e1;
    const int s = pos[2 * b + 0];
    const int e = pos[2 * b + 1];

    const int r0 = chunk * CHUNK_ROWS;
    const int lo = max(r0, s + 1);              // mask: idx > s
    const int hi = min(r0 + CHUNK_ROWS, e);     // mask: idx < e
    if (lo >= hi) return;

    const int t = threadIdx.x;                  // 0..191
    const v4f* __restrict__ base =
        (const v4f*)(hidden + (size_t)b * S_ * H_) + t;

    v4f a0 = {0.f, 0.f, 0.f, 0.f};
    v4f a1 = {0.f, 0.f, 0.f, 0.f};
    v4f a2 = {0.f, 0.f, 0.f, 0.f};
    v4f a3 = {0.f, 0.f, 0.f, 0.f};

    int r = lo;
    for (; r + 4 <= hi; r += 4) {
        v4f v0 = __builtin_nontemporal_load(base + (size_t)(r + 0) * (H_ / 4));
        v4f v1 = __builtin_nontemporal_load(base + (size_t)(r + 1) * (H_ / 4));
        v4f v2 = __builtin_nontemporal_load(base + (size_t)(r + 2) * (H_ / 4));
        v4f v3 = __builtin_nontemporal_load(base + (size_t)(r + 3) * (H_ / 4));
        a0 += v0;
        a1 += v1;
        a2 += v2;
        a3 += v3;
    }
    for (; r < hi; ++r) {
        a0 += __builtin_nontemporal_load(base + (size_t)r * (H_ / 4));
    }
    a0 += a1 + a2 + a3;

    float* dst = accum + (size_t)be * H_ + t * 4;
    __hip_atomic_fetch_add(dst + 0, a0.x, __ATOMIC_RELAXED, __HIP_MEMORY_SCOPE_AGENT);
    __hip_atomic_fetch_add(dst + 1, a0.y, __ATOMIC_RELAXED, __HIP_MEMORY_SCOPE_AGENT);
    __hip_atomic_fetch_add(dst + 2, a0.z, __ATOMIC_RELAXED, __HIP_MEMORY_SCOPE_AGENT);
    __hip_atomic_fetch_add(dst + 3, a0.w, __ATOMIC_RELAXED, __HIP_MEMORY_SCOPE_AGENT);
}

// ------------------------------------------ finalize means + concat into x --
__global__ __launch_bounds__(256) void rbert_build_x(
    const float* __restrict__ hidden,
    const float* __restrict__ cls_vec,
    const int*   __restrict__ e1,
    const int*   __restrict__ e2,
    const float* __restrict__ accum,
    float*       __restrict__ x)        // [B, 3H]
{
    int idx = blockIdx.x * 256 + threadIdx.x;
    if (idx >= B_ * K3H) return;
    int b = idx / K3H;
    int c = idx - b * K3H;

    float val;
    if (c < H_) {
        val = cls_vec[(size_t)b * H_ + c];
    } else {
        int ent = (c < 2 * H_) ? 0 : 1;
        int ch  = c - (ent ? 2 * H_ : H_);
        const int* pos = ent ? e2 : e1;
        int s = pos[2 * b + 0];
        int e = pos[2 * b + 1];
        int cnt = e - s - 1;                       // #(s < idx < e)
        if (cnt > 0)
            val = accum[((size_t)b * 2 + ent) * H_ + ch] / (float)cnt;
        else
            val = hidden[((size_t)b * S_ + s) * H_ + ch];   // fallback row s
    }
    x[idx] = val;
}

// ----------------------------------------------- out = x @ W^T + b (WMMA) ---
// One wave per 16x16 output tile. grid = (8 mtiles, 2 ntiles), 32 threads.
// V_WMMA_F32_16X16X4_F32: A 16x4 (v2f/lane), B 4x16 (v2f/lane), C/D 16x16 (v8f).
// Out-of-range N columns are zeroed via a branchless mask multiply so EXEC
// stays all-ones (WMMA requirement) and the inner loop has no saveexec.
__global__ __launch_bounds__(32) void rbert_gemm_wmma(
    const float* __restrict__ x,      // [B, 3H] row-major
    const float* __restrict__ W,      // [L, 3H] row-major  (B-matrix = W^T)
    const float* __restrict__ bias,   // [L]
    float*       __restrict__ out)    // [B, L] row-major
{
    const int mtile = blockIdx.x;          // 0..7
    const int ntile = blockIdx.y;          // 0..1
    const int lane  = threadIdx.x;         // 0..31
    const int hi    = lane >> 4;           // lane group: K pair select
    const int l15   = lane & 15;

    const int m = mtile * 16 + l15;        // A row owned by this lane
    const int n = ntile * 16 + l15;        // B/D column owned by this lane
    const bool  nvalid = (n < L_);
    const float nmask  = nvalid ? 1.0f : 0.0f;

    // lanes 0-15 supply (K0,K1); lanes 16-31 supply (K2,K3) -> contiguous f2
    const float* __restrict__ xrow = x + (size_t)m * K3H + (hi ? 2 : 0);
    const float* __restrict__ wrow = W + (size_t)(nvalid ? n : 0) * K3H + (hi ? 2 : 0);

    v8f c = {};
    #pragma unroll 4
    for (int k = 0; k < K3H; k += 4) {
        v2f a;  a.x = xrow[k];          a.y = xrow[k + 1];
        v2f bm; bm.x = wrow[k] * nmask; bm.y = wrow[k + 1] * nmask;
        // (neg_a, A, neg_b, B, c_mod, C, reuse_a, reuse_b)
        c = __builtin_amdgcn_wmma_f32_16x16x4_f32(
                false, a, false, bm, (short)0, c, false, false);
    }

    if (nvalid) {
        const float bb = bias[n];
        #pragma unroll
        for (int r = 0; r < 8; ++r) {
            // C/D layout: VGPR r -> M=r (lanes 0-15) / M=r+8 (lanes 16-31)
            int mm = mtile * 16 + r + (hi ? 8 : 0);
            out[(size_t)mm * L_ + n] = c[r] + bb;
        }
    }
}

// ---------------------------------------------------------------------------
extern "C" void kernel_launch(void* const* d_in, const int* in_sizes, int n_in,
                              void* d_out, int out_size, void* d_ws, size_t ws_size,
                              hipStream_t stream) {
    const float* hidden  = (const float*)d_in[0];   // (B,S,H)
    const float* cls_vec = (const float*)d_in[1];   // (B,H)
    const int*   e1      = (const int*)  d_in[2];   // (B,2)
    const int*   e2      = (const int*)  d_in[3];   // (B,2)
    const float* W       = (const float*)d_in[4];   // (L,3H)
    const float* bias    = (const float*)d_in[5];   // (L,)
    float*       out     = (float*)d_out;           // (B,L)

    float* accum = (float*)d_ws;                    // B*2*H floats
    float* x     = accum + (size_t)B_ * 2 * H_;     // B*3H floats

    const int accum_n = B_ * 2 * H_;                // 196608
    rbert_zero<<<(accum_n + 255) / 256, 256, 0, stream>>>(accum, accum_n);

    rbert_span_partial<<<dim3(NCHUNK, B_ * 2), 192, 0, stream>>>(
        hidden, e1, e2, accum);

    const int xn = B_ * K3H;                        // 294912
    rbert_build_x<<<(xn + 255) / 256, 256, 0, stream>>>(
        hidden, cls_vec, e1, e2, accum, x);

    rbert_gemm_wmma<<<dim3(B_ / 16, 2), 32, 0, stream>>>(x, W, bias, out);
}